// Attention_22986664968282
// MI455X (gfx1250) — compile-verified
//
#include <hip/hip_runtime.h>
#include <math.h>

typedef __attribute__((ext_vector_type(2))) float v2f;
typedef __attribute__((ext_vector_type(8))) float v8f;

__device__ __forceinline__ v8f wmma4(v2f a, v2f b, v8f c) {
    // D = A(16x4) * B(4x16) + C(16x16), fp32
    return __builtin_amdgcn_wmma_f32_16x16x4_f32(
        false, a, false, b, (short)0, c, false, false);
}

// --- CDNA5 async global->LDS DMA (ASYNCcnt-tracked), via inline asm ---------
__device__ __forceinline__ void async_ld_b128(unsigned lds, const float* g) {
    asm volatile("global_load_async_to_lds_b128 %0, %1, off"
                 :: "v"(lds), "v"(g) : "memory");
}
__device__ __forceinline__ void async_ld_b32(unsigned lds, const float* g) {
    asm volatile("global_load_async_to_lds_b32 %0, %1, off"
                 :: "v"(lds), "v"(g) : "memory");
}
__device__ __forceinline__ void wait_async0() {
    asm volatile("s_wait_asynccnt 0x0" ::: "memory");
}

// ---------------------------------------------------------------------------
// fp32 WMMA GEMM: C = A[M,K] @ B[K,N], 64x64 block tiles, double-buffered
// K-chunks staged with async-to-LDS DMA.
// MODE 0: scatter into Q/K/V head-major buffers [B*H, P, 64]
// MODE 1: C0[r*N+c] = acc + bias[c]
// ---------------------------------------------------------------------------
#define BM 64
#define BN 64
#define BK 16

template <int MODE>
__global__ __launch_bounds__(256) void gemm_wmma_f32(
    const float* __restrict__ A, const float* __restrict__ B,
    int M, int N, int K,
    float* __restrict__ C0, float* __restrict__ C1, float* __restrict__ C2,
    const float* __restrict__ bias, int P) {
    __shared__ float As[2][BM][BK + 1];  // padded: conflict-free a-frag reads
    __shared__ float Bs[2][BK][BN];

    const int tid  = threadIdx.x;
    const int wave = tid >> 5;
    const int lane = tid & 31;
    const int l16  = lane & 15;
    const int hi   = lane >> 4;
    const int koff = hi * 2;
    const int wm   = (wave >> 1) * 16;
    const int wn   = (wave & 1) * 32;

    const int rowG0 = blockIdx.y * BM;
    const int colG0 = blockIdx.x * BN;

    const unsigned asb = (unsigned)(uintptr_t)&As[0][0][0];
    const unsigned bsb = (unsigned)(uintptr_t)&Bs[0][0][0];

    // per-thread copy assignments
    const int ar = tid >> 2, ak = (tid & 3) << 2;   // A: 64 rows x 16
    const int bk = tid >> 4, bn = (tid & 15) << 2;  // B: 16 rows x 64
    const float* gA = A + (size_t)(rowG0 + ar) * K + ak;
    const float* gB = B + (size_t)bk * N + colG0 + bn;

    auto issue = [&](int k0, int buf) {
        const float* ga = gA + k0;
        const unsigned ab = asb + (unsigned)(buf * (BM * (BK + 1) * 4) +
                                             ar * ((BK + 1) * 4) + ak * 4);
#pragma unroll
        for (int i = 0; i < 4; ++i) async_ld_b32(ab + i * 4, ga + i);
        async_ld_b128(bsb + (unsigned)(buf * (BK * BN * 4) + bk * (BN * 4) + bn * 4),
                      gB + (size_t)k0 * N);
    };

    v8f acc[2] = {};

    issue(0, 0);
    wait_async0();
    __syncthreads();

    int cur = 0;
    for (int k0 = 0; k0 < K; k0 += BK) {
        if (k0 + BK < K) issue(k0 + BK, cur ^ 1);  // prefetch next chunk

#pragma unroll
        for (int kk = 0; kk < 4; ++kk) {
            const int kb = kk * 4 + koff;
            v2f a;
            a.x = As[cur][wm + l16][kb];
            a.y = As[cur][wm + l16][kb + 1];
#pragma unroll
            for (int c = 0; c < 2; ++c) {
                const int n = wn + c * 16 + l16;
                v2f b;
                b.x = Bs[cur][kb][n];
                b.y = Bs[cur][kb + 1][n];
                acc[c] = wmma4(a, b, acc[c]);
            }
        }

        wait_async0();   // my prefetch landed
        __syncthreads(); // everyone done reading cur + everyone's prefetch done
        cur ^= 1;
    }

    // Epilogue. C layout: vgpr j -> row (hi ? j+8 : j), col = l16 (per tile).
#pragma unroll
    for (int c = 0; c < 2; ++c) {
#pragma unroll
        for (int j = 0; j < 8; ++j) {
            const int m   = wm + (hi ? j + 8 : j);
            const int r   = rowG0 + m;
            const int col = colG0 + wn + c * 16 + l16;
            const float v = acc[c][j];
            if (MODE == 0) {
                const int which = col / 768;
                const int d     = col % 768;
                const int h     = d >> 6;
                const int e     = d & 63;
                const int b     = r / P;
                const int p     = r % P;
                float* dst = (which == 0) ? C0 : ((which == 1) ? C1 : C2);
                dst[(((size_t)(b * 12 + h) * P) + p) * 64 + e] = v;
            } else {
                C0[(size_t)r * N + col] = v + bias[col];
            }
        }
    }
}

// ---------------------------------------------------------------------------
// Per-head LayerNorm over hd=64 for Q (then *hd^-0.5) and K, in place.
// ---------------------------------------------------------------------------
__global__ __launch_bounds__(256) void ln_qk(
    float* __restrict__ Q, float* __restrict__ Kb,
    const float* __restrict__ qs, const float* __restrict__ qb,
    const float* __restrict__ ks, const float* __restrict__ kbias,
    int rowsPerTensor) {
    const int wave = threadIdx.x >> 5;
    const int lane = threadIdx.x & 31;
    int idx = blockIdx.x * 8 + wave;
    if (idx >= 2 * rowsPerTensor) return;
    const bool isK = idx >= rowsPerTensor;
    const int row  = isK ? idx - rowsPerTensor : idx;
    float* base = (isK ? Kb : Q) + (size_t)row * 64;

    float2 v = *(const float2*)(base + lane * 2);
    float s = v.x + v.y;
#pragma unroll
    for (int m = 16; m >= 1; m >>= 1) s += __shfl_xor(s, m, 32);
    const float mean = s * (1.0f / 64.0f);
    const float dx = v.x - mean, dy = v.y - mean;
    float vs = dx * dx + dy * dy;
#pragma unroll
    for (int m = 16; m >= 1; m >>= 1) vs += __shfl_xor(vs, m, 32);
    const float inv = rsqrtf(vs * (1.0f / 64.0f) + 1e-6f);
    const float* sc = isK ? ks : qs;
    const float* bi = isK ? kbias : qb;
    const float mul = isK ? 1.0f : 0.125f;  // hd^-0.5
    float o0 = (dx * inv * sc[lane * 2 + 0] + bi[lane * 2 + 0]) * mul;
    float o1 = (dy * inv * sc[lane * 2 + 1] + bi[lane * 2 + 1]) * mul;
    *(float2*)(base + lane * 2) = make_float2(o0, o1);
}

// ---------------------------------------------------------------------------
// Flash attention, one block = 64 Q rows of one (b,h). blockDim = 128.
// KV tiles of 32 rows, double-buffered via async-to-LDS DMA: the next tile
// streams in while WMMA works on the current one. One barrier per iteration.
// ---------------------------------------------------------------------------
#define TK 32

__global__ __launch_bounds__(128) void flash_attn(
    const float* __restrict__ Q, const float* __restrict__ Kb,
    const float* __restrict__ V, float* __restrict__ O, int P) {
    __shared__ float Qs[64 * 64];       // 16 KB
    __shared__ float Ks[2][TK * 64];    // 16 KB
    __shared__ float Vs[2][TK * 64];    // 16 KB
    __shared__ float Ps[64 * TK];       //  8 KB (wave-private rows)

    const int tid  = threadIdx.x;
    const int wave = tid >> 5;
    const int lane = tid & 31;
    const int l16  = lane & 15;
    const int hi   = lane >> 4;
    const int koff = hi * 2;
    const int qr   = wave * 16 + l16;

    const int bh = blockIdx.y;
    const int q0 = blockIdx.x * 64;

    const unsigned qsb = (unsigned)(uintptr_t)&Qs[0];
    const unsigned ksb = (unsigned)(uintptr_t)&Ks[0][0];
    const unsigned vsb = (unsigned)(uintptr_t)&Vs[0][0];

    const float* Qg  = Q + ((size_t)bh * P + q0) * 64;
    const float* Kg0 = Kb + (size_t)bh * P * 64;
    const float* Vg0 = V + (size_t)bh * P * 64;

    // Q tile: contiguous 64x64 -> 1024 b128 transfers, 8 per thread.
#pragma unroll
    for (int p = 0; p < 8; ++p) {
        const int i = tid + p * 128;
        async_ld_b128(qsb + i * 16, Qg + i * 4);
    }
    auto issueKV = [&](int k0, int buf) {
#pragma unroll
        for (int p = 0; p < 4; ++p) {
            const int i = tid + p * 128;  // 512 b128 per tile
            async_ld_b128(ksb + buf * (TK * 64 * 4) + i * 16,
                          Kg0 + (size_t)k0 * 64 + i * 4);
            async_ld_b128(vsb + buf * (TK * 64 * 4) + i * 16,
                          Vg0 + (size_t)k0 * 64 + i * 4);
        }
    };
    issueKV(0, 0);

    v8f o[4] = {};
    float mrow[8], lrow[8];
#pragma unroll
    for (int j = 0; j < 8; ++j) { mrow[j] = -__builtin_inff(); lrow[j] = 0.0f; }

    wait_async0();
    __syncthreads();

    int cur = 0;
    for (int k0 = 0; k0 < P; k0 += TK) {
        if (k0 + TK < P) issueKV(k0 + TK, cur ^ 1);  // overlap with compute

        // S = Q @ K^T  (K-dim = hd = 64), 2 tiles of 16 KV rows
        v8f s[2] = {};
#pragma unroll
        for (int ks = 0; ks < 16; ++ks) {
            const int kb = ks * 4 + koff;
            v2f a;
            a.x = Qs[qr * 64 + kb];
            a.y = Qs[qr * 64 + kb + 1];
#pragma unroll
            for (int t = 0; t < 2; ++t) {
                v2f b;  // B = K^T: B[k][n] = Ktile[n][k]
                b.x = Ks[cur][(t * 16 + l16) * 64 + kb];
                b.y = Ks[cur][(t * 16 + l16) * 64 + kb + 1];
                s[t] = wmma4(a, b, s[t]);
            }
        }

        // Online softmax; stage P in wave-private rows of Ps.
#pragma unroll
        for (int j = 0; j < 8; ++j) {
            float mx = fmaxf(s[0][j], s[1][j]);
#pragma unroll
            for (int m = 1; m <= 8; m <<= 1) mx = fmaxf(mx, __shfl_xor(mx, m, 32));
            const float mnew = fmaxf(mrow[j], mx);
            const float f = __expf(mrow[j] - mnew);
            mrow[j] = mnew;
            const int prow = wave * 16 + (hi ? j + 8 : j);
            float psum = 0.0f;
#pragma unroll
            for (int t = 0; t < 2; ++t) {
                const float p = __expf(s[t][j] - mnew);
                psum += p;
                Ps[prow * TK + t * 16 + l16] = p;
            }
#pragma unroll
            for (int m = 1; m <= 8; m <<= 1) psum += __shfl_xor(psum, m, 32);
            lrow[j] = lrow[j] * f + psum;
#pragma unroll
            for (int t = 0; t < 4; ++t) o[t][j] *= f;
        }

        // O += P @ V  (K-dim = TK KV rows)
#pragma unroll
        for (int ks = 0; ks < 8; ++ks) {
            const int kb = ks * 4 + koff;
            v2f a;
            a.x = Ps[qr * TK + kb];
            a.y = Ps[qr * TK + kb + 1];
#pragma unroll
            for (int d = 0; d < 4; ++d) {
                v2f b;
                b.x = Vs[cur][(kb + 0) * 64 + d * 16 + l16];
                b.y = Vs[cur][(kb + 1) * 64 + d * 16 + l16];
                o[d] = wmma4(a, b, o[d]);
            }
        }

        wait_async0();   // next KV tile landed
        __syncthreads(); // all reads of cur done everywhere
        cur ^= 1;
    }

    // Normalize and write merged-head layout [B, P, D].
    const int b = bh / 12, h = bh % 12;
#pragma unroll
    for (int j = 0; j < 8; ++j) {
        const int prow = q0 + wave * 16 + (hi ? j + 8 : j);
        const float invl = 1.0f / lrow[j];
#pragma unroll
        for (int d = 0; d < 4; ++d) {
            O[(size_t)(b * P + prow) * 768 + h * 64 + d * 16 + l16] =
                o[d][j] * invl;
        }
    }
}

// ---------------------------------------------------------------------------
// LayerNorm over D=768, one wave per row.
// ---------------------------------------------------------------------------
__global__ __launch_bounds__(256) void ln_out(
    const float* __restrict__ X, float* __restrict__ Y,
    const float* __restrict__ sc, const float* __restrict__ bi) {
    const int wave = threadIdx.x >> 5;
    const int lane = threadIdx.x & 31;
    const int row  = blockIdx.x * 8 + wave;
    const float* x = X + (size_t)row * 768;

    float v[24];
    float s = 0.0f;
#pragma unroll
    for (int i = 0; i < 24; ++i) { v[i] = x[lane + i * 32]; s += v[i]; }
#pragma unroll
    for (int m = 16; m >= 1; m >>= 1) s += __shfl_xor(s, m, 32);
    const float mean = s * (1.0f / 768.0f);
    float var = 0.0f;
#pragma unroll
    for (int i = 0; i < 24; ++i) { v[i] -= mean; var += v[i] * v[i]; }
#pragma unroll
    for (int m = 16; m >= 1; m >>= 1) var += __shfl_xor(var, m, 32);
    const float inv = rsqrtf(var * (1.0f / 768.0f) + 1e-6f);
    float* y = Y + (size_t)row * 768;
#pragma unroll
    for (int i = 0; i < 24; ++i) {
        const int e = lane + i * 32;
        y[e] = v[i] * inv * sc[e] + bi[e];
    }
}

// ---------------------------------------------------------------------------
extern "C" void kernel_launch(void* const* d_in, const int* in_sizes, int n_in,
                              void* d_out, int out_size, void* d_ws,
                              size_t ws_size, hipStream_t stream) {
    const float* x       = (const float*)d_in[0];
    const float* W_qkv   = (const float*)d_in[1];
    const float* q_scale = (const float*)d_in[2];
    const float* q_bias  = (const float*)d_in[3];
    const float* k_scale = (const float*)d_in[4];
    const float* k_bias  = (const float*)d_in[5];
    const float* o_scale = (const float*)d_in[6];
    const float* o_bias  = (const float*)d_in[7];
    const float* W_out   = (const float*)d_in[8];
    const float* b_out   = (const float*)d_in[9];
    float* out = (float*)d_out;
    float* ws  = (float*)d_ws;

    const int B = 2, P = 2048, H = 12;
    const int Mrows = B * P;                    // 4096
    const size_t seg = (size_t)B * H * P * 64;  // floats per Q/K/V buffer

    float* Qb = ws;
    float* Kb = ws + seg;
    float* Vb = ws + 2 * seg;
    float* X2 = ws + 3 * seg;  // attention output, merged heads [B*P, 768]
    float* Y  = Qb;            // LN output reuses Q segment (Q dead by then)

    // 1) QKV projection: [4096,768] @ [768,2304] -> head-major Q/K/V
    {
        dim3 grid(2304 / BN, Mrows / BM);
        gemm_wmma_f32<0><<<grid, 256, 0, stream>>>(
            x, W_qkv, Mrows, 2304, 768, Qb, Kb, Vb, nullptr, P);
    }
    // 2) Per-head LayerNorm on Q (with *hd^-0.5) and K
    {
        const int rows = B * H * P;
        ln_qk<<<(2 * rows) / 8, 256, 0, stream>>>(
            Qb, Kb, q_scale, q_bias, k_scale, k_bias, rows);
    }
    // 3) Flash attention per (b,h)
    {
        dim3 grid(P / 64, B * H);
        flash_attn<<<grid, 128, 0, stream>>>(Qb, Kb, Vb, X2, P);
    }
    // 4) Output LayerNorm over D=768
    ln_out<<<Mrows / 8, 256, 0, stream>>>(X2, Y, o_scale, o_bias);
    // 5) Final projection: [4096,768] @ [768,768] + b_out -> d_out
    {
        dim3 grid(768 / BN, Mrows / BM);
        gemm_wmma_f32<1><<<grid, 256, 0, stream>>>(
            Y, W_out, Mrows, 768, 768, out, nullptr, nullptr, b_out, P);
    }
}